// CentroidTripletLoss_46883863003641
// MI455X (gfx1250) — compile-verified
//
#include <hip/hip_runtime.h>

typedef __attribute__((ext_vector_type(2))) float v2f;
typedef __attribute__((ext_vector_type(8))) float v8f;

#define N_ROWS 8192
#define DIM    2048
#define NCLS   256
#define MARGIN 0.3f

// ---------------------------------------------------------------------------
// Kernel 1: centroids via ordered compaction.
// Block = (col-chunk x, class y), 256 threads.
// Phase A: each thread scans a 32-row strip for label==c, prefix-sum counts,
//          compact matching row ids into LDS in ascending row order
//          (deterministic accumulation order == reference segment order).
// Phase B: loop over the ~32 matches; coalesced 256-float row-slice loads.
// ---------------------------------------------------------------------------
__global__ __launch_bounds__(256) void centroid_kernel(
    const float* __restrict__ E, const int* __restrict__ labels,
    float* __restrict__ cent) {
  const int c   = blockIdx.y;
  const int t   = threadIdx.x;
  const int col = blockIdx.x * 256 + t;

  __shared__ int rows_l[N_ROWS];   // worst-case all rows in one class (32KB)
  __shared__ int pre[256];

  // Phase A1: count matches in my strip (rows [t*32, t*32+32))
  int myCnt = 0;
  for (int j = 0; j < 32; ++j) {
    if (labels[t * 32 + j] == c) ++myCnt;
  }
  pre[t] = myCnt;
  __syncthreads();

  // Inclusive prefix sum (Hillis-Steele)
  for (int s = 1; s < 256; s <<= 1) {
    int v = (t >= s) ? pre[t - s] : 0;
    __syncthreads();
    pre[t] += v;
    __syncthreads();
  }
  const int base  = pre[t] - myCnt;   // exclusive offset of my strip
  const int total = pre[255];

  // Phase A2: compact matching row ids in order
  int k = 0;
  for (int j = 0; j < 32; ++j) {
    int row = t * 32 + j;
    if (labels[row] == c) rows_l[base + k++] = row;
  }
  __syncthreads();

  // Phase B: accumulate my column over the matched rows (ascending order)
  float acc = 0.0f;
  for (int i = 0; i < total; ++i) {
    acc += E[(size_t)rows_l[i] * DIM + col];
  }
  cent[(size_t)c * DIM + col] = (total > 0) ? (acc / (float)total) : 0.0f;
}

// ---------------------------------------------------------------------------
// Kernel 2: fused WMMA GEMM + norms + distance + per-row ap/an/hinge.
// Block = 512 threads = 16 waves; block owns 16 rows, waves cover all 256 cols.
// fp32 WMMA 16x16x4 per ISA 7.12.2 layouts:
//   A: lanes 0-15 K=0/1 (VGPR0/1), lanes 16-31 K=2/3; B mirrored (N = lane&15)
//   D: VGPR v -> row v+8*half, col = lane&15.
// ||e||^2 / ||c||^2 are folded into the K loop from the same registers.
// The 16x256 dist tile lives in LDS; wave w reduces row w -> rowloss.
// ---------------------------------------------------------------------------
__global__ __launch_bounds__(512) void gemm_loss_kernel(
    const float* __restrict__ E, const float* __restrict__ cent,
    const int* __restrict__ labels, float* __restrict__ rowloss) {
  const int wave = threadIdx.x >> 5;          // 0..15
  const int lane = threadIdx.x & 31;
  const int half = lane >> 4;                 // 0 or 1
  const int m    = lane & 15;

  const int rowBase = blockIdx.x * 16;
  const int colBase = wave * 16;

  const float* aPtr = E    + (size_t)(rowBase + m) * DIM + 2 * half;
  const float* bPtr = cent + (size_t)(colBase + m) * DIM + 2 * half;

  v8f acc = {};
  float eSq = 0.0f, cSq = 0.0f;
#pragma unroll 8
  for (int kk = 0; kk < DIM; kk += 4) {
    v2f a = *(const v2f*)(aPtr + kk);
    v2f b = *(const v2f*)(bPtr + kk);
    eSq += a.x * a.x + a.y * a.y;             // partial ||e_row||^2 (this K half)
    cSq += b.x * b.x + b.y * b.y;             // partial ||c_col||^2 (this K half)
    acc = __builtin_amdgcn_wmma_f32_16x16x4_f32(
        /*neg_a=*/false, a, /*neg_b=*/false, b,
        /*c_mod=*/(short)0, acc, /*reuse_a=*/false, /*reuse_b=*/false);
  }
  // Combine the two K-halves held by lanes m and m+16
  const float eTot = eSq + __shfl_xor(eSq, 16, 32);   // ||e_{rowBase+(lane&15)}||^2
  const float cTot = cSq + __shfl_xor(cSq, 16, 32);   // ||c_{colBase+(lane&15)}||^2

  __shared__ float tile[16][NCLS + 1];        // +1 pad: conflict-free banks
  const int col = colBase + m;
#pragma unroll
  for (int v = 0; v < 8; ++v) {
    const int r = v + 8 * half;
    float er = __shfl(eTot, r, 32);           // ||e||^2 of row r (bpermute)
    float d2 = er + cTot - 2.0f * acc[v];
    d2 = d2 > 0.0f ? d2 : 0.0f;
    tile[r][col] = sqrtf(d2);
  }
  __syncthreads();

  // Wave w reduces row w of the tile: an = min over cols != label, ap = tile[lab]
  const int lab = labels[rowBase + wave];     // uniform per wave -> s_load
  float mn = __builtin_inff();
#pragma unroll
  for (int i = 0; i < NCLS / 32; ++i) {
    const int cc = lane + 32 * i;
    const float dv = tile[wave][cc];
    mn = (cc == lab) ? mn : fminf(mn, dv);
  }
  for (int off = 16; off > 0; off >>= 1)
    mn = fminf(mn, __shfl_xor(mn, off, 32));
  if (lane == 0) {
    const float ap = tile[wave][lab];
    const float v  = ap - mn + MARGIN;
    rowloss[rowBase + wave] = v > 0.0f ? v : 0.0f;
  }
}

// ---------------------------------------------------------------------------
// Kernel 3: mean over rows -> out[0]  (single block; deterministic, no atomics)
// ---------------------------------------------------------------------------
__global__ __launch_bounds__(256) void mean_kernel(
    const float* __restrict__ rowloss, float* __restrict__ out) {
  const int t = threadIdx.x;
  float p = 0.0f;
  for (int i = 0; i < N_ROWS / 256; ++i) p += rowloss[t + 256 * i];
  __shared__ float red[256];
  red[t] = p;
  __syncthreads();
  for (int s = 128; s > 0; s >>= 1) {
    if (t < s) red[t] += red[t + s];
    __syncthreads();
  }
  if (t == 0) out[0] = red[0] * (1.0f / (float)N_ROWS);
}

// ---------------------------------------------------------------------------
extern "C" void kernel_launch(void* const* d_in, const int* in_sizes, int n_in,
                              void* d_out, int out_size, void* d_ws, size_t ws_size,
                              hipStream_t stream) {
  const float* E      = (const float*)d_in[0];   // [8192, 2048] f32
  const int*   labels = (const int*)d_in[1];     // [8192] int32
  float*       out    = (float*)d_out;           // scalar f32

  // Workspace layout (floats): cent (2MB) + rowloss (32KB)
  float* ws      = (float*)d_ws;
  float* cent    = ws;                            // 256*2048
  float* rowloss = cent + (size_t)NCLS * DIM;     // 8192

  centroid_kernel<<<dim3(DIM / 256, NCLS), 256, 0, stream>>>(E, labels, cent);
  gemm_loss_kernel<<<dim3(N_ROWS / 16), 512, 0, stream>>>(E, cent, labels, rowloss);
  mean_kernel<<<1, 256, 0, stream>>>(rowloss, out);
}